// CompGCNLayer_15204184228262
// MI455X (gfx1250) — compile-verified
//
#include <hip/hip_runtime.h>

#define DDIM 128
#define LN_EPS 1e-5f
#define LDS_STRIDE 132   // 128 + 4 pad -> consecutive rows land on different banks

typedef __attribute__((ext_vector_type(2))) float v2f;
typedef __attribute__((ext_vector_type(8))) float v8f;

// ---------------- zero the scatter accumulator ----------------
__global__ void zero_f4_kernel(float4* __restrict__ p, long n4) {
    long i = (long)blockIdx.x * blockDim.x + threadIdx.x;
    if (i < n4) p[i] = make_float4(0.f, 0.f, 0.f, 0.f);
}

// ---------------- edge gather * rel, scatter-add --------------
// one wave32 per edge; lane handles a float4 chunk of D=128
__global__ __launch_bounds__(256)
void edge_scatter_kernel(const float* __restrict__ ent,
                         const float* __restrict__ rel,
                         const int*   __restrict__ eidx,   // [2, E] flat
                         const int*   __restrict__ etype,  // [E]
                         float* __restrict__ agg, int E) {
    int wave = (int)((blockIdx.x * blockDim.x + threadIdx.x) >> 5);
    int lane = threadIdx.x & 31;
    if (wave >= E) return;
    int src = eidx[wave];
    int dst = eidx[E + wave];
    int r   = etype[wave];

    const float4 a = *(const float4*)(ent + (long)src * DDIM + lane * 4);
    const float4 m = *(const float4*)(rel + (long)r   * DDIM + lane * 4);
    float* o = agg + (long)dst * DDIM + lane * 4;
    // hardware FP32 atomics (global_atomic_add_f32) into L2
    unsafeAtomicAdd(o + 0, a.x * m.x);
    unsafeAtomicAdd(o + 1, a.y * m.y);
    unsafeAtomicAdd(o + 2, a.z * m.z);
    unsafeAtomicAdd(o + 3, a.w * m.w);
}

// ------- h = agg @ W^T + b; out = LN(h + ent) fused, WMMA fp32 -------
// block = 256 threads = 8 waves; block owns 16 rows; wave w owns cols [16w,16w+16)
__global__ __launch_bounds__(256)
void gemm_ln_kernel(const float* __restrict__ agg,
                    const float* __restrict__ W,      // [D,D] row-major; h[n,j]=sum_k agg[n,k]*W[j,k]
                    const float* __restrict__ bias,
                    const float* __restrict__ gamma,
                    const float* __restrict__ beta,
                    const float* __restrict__ ent,
                    float* __restrict__ out, int N) {
    __shared__ float As[16 * LDS_STRIDE];
    __shared__ float Hs[16 * LDS_STRIDE];

    const int tid  = threadIdx.x;
    const int wv   = tid >> 5;        // 0..7
    const int lane = tid & 31;
    const long row0 = (long)blockIdx.x * 16;

    // cooperative load of the 16x128 A tile (512 float4 loads / 256 threads)
    for (int i = tid; i < 16 * 32; i += 256) {
        int  rr = i >> 5;
        int  cc = (i & 31) * 4;
        long gr = row0 + rr; if (gr >= N) gr = N - 1;   // clamp (N is a multiple of 16 anyway)
        float4 v = *(const float4*)(agg + gr * DDIM + cc);
        float* d = &As[rr * LDS_STRIDE + cc];
        d[0] = v.x; d[1] = v.y; d[2] = v.z; d[3] = v.w;
    }
    __syncthreads();

    const int col0 = wv * 16;
    const int n    = lane & 15;            // M for A, N for B/C
    const int koff = (lane >> 4) * 2;      // lanes 16-31 hold K+2 / K+3
    const float* wrow = W + (long)(col0 + n) * DDIM;  // B[k][n] = W[col0+n][k]

    v8f c = {};
#pragma unroll
    for (int k0 = 0; k0 < DDIM; k0 += 4) {
        v2f a, bb;
        a[0]  = As[n * LDS_STRIDE + k0 + koff];
        a[1]  = As[n * LDS_STRIDE + k0 + koff + 1];
        bb[0] = wrow[k0 + koff];
        bb[1] = wrow[k0 + koff + 1];
        // D(16x16,f32) += A(16x4,f32) x B(4x16,f32)  -> v_wmma_f32_16x16x4_f32
        c = __builtin_amdgcn_wmma_f32_16x16x4_f32(false, a, false, bb,
                                                  (short)0, c, false, false);
    }

    // C layout: VGPR r -> (M=r, N=lane) lanes 0-15 ; (M=r+8, N=lane-16) lanes 16-31
    {
        const int cc = col0 + n;
        const float bv = bias[cc];
#pragma unroll
        for (int r = 0; r < 8; ++r) {
            int rr = r + (lane >> 4) * 8;
            Hs[rr * LDS_STRIDE + cc] = c[r] + bv;
        }
    }
    __syncthreads();

    // fused residual + LayerNorm: wave wv handles rows 2wv, 2wv+1
    for (int rr = 2 * wv; rr < 2 * wv + 2; ++rr) {
        long grow = row0 + rr;
        long gclmp = grow >= N ? N - 1 : grow;
        float x[4];
        float s = 0.f;
#pragma unroll
        for (int i = 0; i < 4; ++i) {
            int cc = lane * 4 + i;
            x[i] = Hs[rr * LDS_STRIDE + cc] + ent[gclmp * DDIM + cc];
            s += x[i];
        }
#pragma unroll
        for (int m = 16; m > 0; m >>= 1) s += __shfl_xor(s, m, 32);
        float mu = s * (1.0f / DDIM);
        float sq = 0.f;
#pragma unroll
        for (int i = 0; i < 4; ++i) { float d = x[i] - mu; sq += d * d; }
#pragma unroll
        for (int m = 16; m > 0; m >>= 1) sq += __shfl_xor(sq, m, 32);
        float rstd = rsqrtf(sq * (1.0f / DDIM) + LN_EPS);
        if (grow < N) {
#pragma unroll
            for (int i = 0; i < 4; ++i) {
                int cc = lane * 4 + i;
                out[grow * DDIM + cc] = (x[i] - mu) * rstd * gamma[cc] + beta[cc];
            }
        }
    }
}

extern "C" void kernel_launch(void* const* d_in, const int* in_sizes, int n_in,
                              void* d_out, int out_size, void* d_ws, size_t ws_size,
                              hipStream_t stream) {
    const float* ent   = (const float*)d_in[0];
    const float* rel   = (const float*)d_in[1];
    const int*   eidx  = (const int*)  d_in[2];
    const int*   etype = (const int*)  d_in[3];
    const float* W     = (const float*)d_in[4];
    const float* bias  = (const float*)d_in[5];
    const float* gamma = (const float*)d_in[6];
    const float* beta  = (const float*)d_in[7];
    float* out = (float*)d_out;

    const int N = in_sizes[0] / DDIM;   // 100000
    const int R = in_sizes[1] / DDIM;   // 500
    const int E = in_sizes[3];          // 600000

    float* agg = (float*)d_ws;          // [N, D] fp32 accumulator

    long n4 = (long)N * DDIM / 4;
    zero_f4_kernel<<<(int)((n4 + 255) / 256), 256, 0, stream>>>((float4*)agg, n4);

    edge_scatter_kernel<<<(E + 7) / 8, 256, 0, stream>>>(ent, rel, eidx, etype, agg, E);

    gemm_ln_kernel<<<(N + 15) / 16, 256, 0, stream>>>(agg, W, bias, gamma, beta, ent, out, N);

    // tuple output: append unchanged rel_emb after out[N*D]
    hipMemcpyAsync(out + (long)N * DDIM, rel, (size_t)R * DDIM * sizeof(float),
                   hipMemcpyDeviceToDevice, stream);
}